// GNNEncoder_13134009991763
// MI455X (gfx1250) — compile-verified
//
#include <hip/hip_runtime.h>

typedef __attribute__((ext_vector_type(2))) float v2f;
typedef __attribute__((ext_vector_type(4))) float v4f;
typedef __attribute__((ext_vector_type(8))) float v8f;
typedef __attribute__((ext_vector_type(4))) int v4i;

#define LDST 260  // LDS row stride in floats (bank-conflict-free column reads)
#define AS1 __attribute__((address_space(1)))
#define AS3 __attribute__((address_space(3)))

__device__ __forceinline__ v8f wmma_f32(v2f a, v2f b, v8f c) {
  // D = A(16x4,f32) * B(4x16,f32) + C(16x16,f32) -> v_wmma_f32_16x16x4_f32
  return __builtin_amdgcn_wmma_f32_16x16x4_f32(false, a, false, b, (short)0, c,
                                               false, false);
}

// Pack W[K x N] into row-pair-interleaved Wp so a B fragment is one b64 load:
// Wp[((k>>1)*N + c)*2 + (k&1)] = W[k*N + c]
__global__ __launch_bounds__(256) void pack_pairs(const float* __restrict__ W,
                                                  float* __restrict__ Wp, int K,
                                                  int N) {
  const int idx = blockIdx.x * 256 + threadIdx.x;
  if (idx < K * N) {
    const int k = idx / N, c = idx % N;
    Wp[((k >> 1) * N + c) * 2 + (k & 1)] = W[idx];
  }
}

// GEMM: hs[64 x NOUT] = xs[64 x K] @ W[K x NOUT] (xs,hs LDS; Wp packed global).
// Each wave owns a NOUT/8-wide column slab across all four 16-row blocks, so
// every W element is loaded by exactly one wave per block.
template <int K, int NOUT>
__device__ __forceinline__ void gemm_lds(const float* xs,
                                         const v2f* __restrict__ Wp, float* hs,
                                         int tid) {
  const int wave = tid >> 5;
  const int lane = tid & 31;
  const int half = lane >> 4;  // lanes 16-31 carry K+2 (A/B), M+8 (C/D)
  const int m = lane & 15;
  constexpr int NT = NOUT / 128;  // N tiles per wave: 2 (NOUT=256) or 1 (128)
  const int nb0 = wave * (NOUT / 8);

  v8f acc[4][NT];
#pragma unroll
  for (int mt = 0; mt < 4; ++mt)
#pragma unroll
    for (int t = 0; t < NT; ++t)
#pragma unroll
      for (int r = 0; r < 8; ++r) acc[mt][t][r] = 0.0f;

  for (int k0 = 0; k0 < K; k0 += 4) {
    const int ka = k0 + 2 * half;        // A/B rows this half-wave covers
    const int p = (k0 >> 1) + half;      // packed row-pair index
    v2f a[4];
#pragma unroll
    for (int mt = 0; mt < 4; ++mt)       // ds_load_b64, conflict-free
      a[mt] = *(const v2f*)&xs[(mt * 16 + m) * LDST + ka];
    v2f b[NT];
#pragma unroll
    for (int t = 0; t < NT; ++t)         // one coalesced global_load_b64
      b[t] = Wp[p * NOUT + nb0 + t * 16 + m];
#pragma unroll
    for (int mt = 0; mt < 4; ++mt)
#pragma unroll
      for (int t = 0; t < NT; ++t)
        acc[mt][t] = wmma_f32(a[mt], b[t], acc[mt][t]);
  }
#pragma unroll
  for (int mt = 0; mt < 4; ++mt)
#pragma unroll
    for (int t = 0; t < NT; ++t) {
      const int col = nb0 + t * 16 + m;
#pragma unroll
      for (int r = 0; r < 8; ++r)  // C/D: VGPR r -> row r (lanes<16) / r+8
        hs[(mt * 16 + r + 8 * half) * LDST + col] = acc[mt][t][r];
    }
  __syncthreads();
}

// GCN aggregation in LDS: xs[n] = relu(sum_{e:dst==n} norm[e]*hs[src] +
//                                      dinv[n]^2*hs[n] + bias)
template <int DOUT>
__device__ __forceinline__ void aggregate(const float* __restrict__ bias,
                                          float* xs, const float* hs,
                                          const int* srcL, const int* dstL,
                                          const float* enorm,
                                          const float* dinv_s, int tid) {
  const int wave = tid >> 5;
  const int lane = tid & 31;
  constexpr int C4 = DOUT / 4;
  // self-loop init (vectorized)
  for (int idx = tid; idx < 64 * C4; idx += 256) {
    const int n = idx / C4, c = (idx % C4) * 4;
    const float s = dinv_s[n] * dinv_s[n];
    v4f h4 = *(const v4f*)&hs[n * LDST + c];
    *(v4f*)&xs[n * LDST + c] = s * h4;
  }
  __syncthreads();
  // edge scatter: one wave per edge -> edge scalars are wave-uniform
  for (int e = wave; e < 128; e += 8) {
    const int s = srcL[e], d = dstL[e];
    const float w = enorm[e];
#pragma unroll
    for (int j = 0; j < DOUT / 32; ++j) {
      const int c = lane + 32 * j;
      atomicAdd(&xs[d * LDST + c], w * hs[s * LDST + c]);  // ds_add_f32
    }
  }
  __syncthreads();
  // bias + relu (vectorized)
  for (int idx = tid; idx < 64 * C4; idx += 256) {
    const int n = idx / C4, c = (idx % C4) * 4;
    v4f v = *(const v4f*)&xs[n * LDST + c];
    v4f bb = *(const v4f*)&bias[c];
    v4f r;
#pragma unroll
    for (int q = 0; q < 4; ++q) r[q] = fmaxf(v[q] + bb[q], 0.0f);
    *(v4f*)&xs[n * LDST + c] = r;
  }
  __syncthreads();
}

// h1 = qubit_embs(64x128) @ W1(128x256), computed once, shared by all graphs.
__global__ __launch_bounds__(256) void h1_kernel(const float* __restrict__ Q,
                                                 const float* __restrict__ W1,
                                                 float* __restrict__ h1) {
  const int tid = threadIdx.x;
  const int wave = tid >> 5, lane = tid & 31, half = lane >> 4, m = lane & 15;
  const int mb = (wave >> 1) * 16;
  const int nb0 = (wave & 1) * 128;
  v8f acc[8];
#pragma unroll
  for (int t = 0; t < 8; ++t)
#pragma unroll
    for (int r = 0; r < 8; ++r) acc[t][r] = 0.0f;

  for (int k0 = 0; k0 < 128; k0 += 4) {
    const int ka = k0 + 2 * half;
    v2f a;
    a.x = Q[(mb + m) * 128 + ka];
    a.y = Q[(mb + m) * 128 + ka + 1];
#pragma unroll
    for (int t = 0; t < 8; ++t) {
      const int col = nb0 + t * 16 + m;
      v2f b;
      b.x = W1[ka * 256 + col];
      b.y = W1[(ka + 1) * 256 + col];
      acc[t] = wmma_f32(a, b, acc[t]);
    }
  }
#pragma unroll
  for (int t = 0; t < 8; ++t) {
    const int col = nb0 + t * 16 + m;
#pragma unroll
    for (int r = 0; r < 8; ++r)
      h1[(mb + r + 8 * half) * 256 + col] = acc[t][r];
  }
}

// One workgroup per graph: whole 3-layer GCN fused in LDS.
__global__ __launch_bounds__(256) void gcn_fused(
    const float* __restrict__ h1, const float* __restrict__ b1,
    const v2f* __restrict__ Wp2, const float* __restrict__ b2,
    const v2f* __restrict__ Wp3, const float* __restrict__ b3,
    const long long* __restrict__ edge, long long totalE,
    float* __restrict__ out) {
  __shared__ float xs[64 * LDST];
  __shared__ float hs[64 * LDST];
  __shared__ int srcL[128];
  __shared__ int dstL[128];
  __shared__ float enorm[128];
  __shared__ float dinv_s[64];
  __shared__ float degs[64];

  const int tid = threadIdx.x;
  const long long g = blockIdx.x;

  // ---- degrees + symmetric normalization (matches reference) ----
  if (tid < 128) {
    const long long s = edge[g * 128 + tid];
    const long long d = edge[totalE + g * 128 + tid];
    srcL[tid] = (int)(s & 63);  // src = g*64 + local
    dstL[tid] = (int)(d & 63);
  }
  if (tid < 64) degs[tid] = 1.0f;  // self loop
  __syncthreads();
  if (tid < 128) atomicAdd(&degs[dstL[tid]], 1.0f);  // ds_add_f32
  __syncthreads();
  if (tid < 64) dinv_s[tid] = rsqrtf(degs[tid]);  // deg >= 1 always
  __syncthreads();
  if (tid < 128) enorm[tid] = dinv_s[srcL[tid]] * dinv_s[dstL[tid]];

  // ---- layer 1 input: h1 shared across graphs (x = tile(qubit_embs)) ----
  {
    const int n = tid >> 2, q = tid & 3;  // row, 64-float chunk
    const float* gsrc = h1 + n * 256 + q * 64;
    float* ldst = &hs[n * LDST + q * 64];
#if __has_builtin(__builtin_amdgcn_global_load_async_to_lds_b128)
#pragma unroll
    for (int j = 0; j < 16; ++j)
      __builtin_amdgcn_global_load_async_to_lds_b128(
          (AS1 v4i*)(gsrc + j * 4), (AS3 v4i*)(ldst + j * 4), 0, 0);
    asm volatile("s_wait_asynccnt 0x0" ::: "memory");
#else
#pragma unroll
    for (int j = 0; j < 16; ++j)
      *(v4f*)(ldst + j * 4) = *(const v4f*)(gsrc + j * 4);
#endif
  }
  __syncthreads();
  aggregate<256>(b1, xs, hs, srcL, dstL, enorm, dinv_s, tid);

  // ---- layer 2 ----
  gemm_lds<256, 256>(xs, Wp2, hs, tid);
  aggregate<256>(b2, xs, hs, srcL, dstL, enorm, dinv_s, tid);

  // ---- layer 3 ----
  gemm_lds<256, 128>(xs, Wp3, hs, tid);
  aggregate<128>(b3, xs, hs, srcL, dstL, enorm, dinv_s, tid);

  // ---- streaming (non-temporal) result write ----
  for (int idx = tid; idx < 64 * 32; idx += 256) {
    const int n = idx >> 5, c = (idx & 31) * 4;
    v4f v = *(const v4f*)&xs[n * LDST + c];
    __builtin_nontemporal_store(v, (v4f*)&out[(g * 64 + n) * 128 + c]);
  }
}

extern "C" void kernel_launch(void* const* d_in, const int* in_sizes, int n_in,
                              void* d_out, int out_size, void* d_ws,
                              size_t ws_size, hipStream_t stream) {
  const float* Q  = (const float*)d_in[0];
  const float* W1 = (const float*)d_in[1];
  const float* b1 = (const float*)d_in[2];
  const float* W2 = (const float*)d_in[3];
  const float* b2 = (const float*)d_in[4];
  const float* W3 = (const float*)d_in[5];
  const float* b3 = (const float*)d_in[6];
  const long long* edge = (const long long*)d_in[7];  // int64 [2, B*E]

  const long long totalE = (long long)in_sizes[7] / 2;  // B*E
  const int numGraphs = (int)(totalE / 128);

  float* h1  = (float*)d_ws;        // 64*256   = 16384 floats
  float* Wp2 = h1 + 16384;          // 256*256  = 65536 floats (packed)
  float* Wp3 = Wp2 + 65536;         // 256*128  = 32768 floats (packed)

  pack_pairs<<<(65536 + 255) / 256, 256, 0, stream>>>(W2, Wp2, 256, 256);
  pack_pairs<<<(32768 + 255) / 256, 256, 0, stream>>>(W3, Wp3, 256, 128);
  h1_kernel<<<1, 256, 0, stream>>>(Q, W1, h1);
  gcn_fused<<<numGraphs, 256, 0, stream>>>(h1, b1, (const v2f*)Wp2, b2,
                                           (const v2f*)Wp3, b3, edge, totalE,
                                           (float*)d_out);
}